// RoutingCaps_42107859370300
// MI455X (gfx1250) — compile-verified
//
#include <hip/hip_runtime.h>
#include <math.h>

// ---------------------------------------------------------------------------
// RoutingCaps on MI455X (gfx1250, wave32).
//
// Identity: sum_z C[b,k,i,z] = 1 + sum_z bias[k,i,z] (softmax row-sum == 1),
// so S[b,k,l] = sum_i U[b,k,i,l] * (1 + bsum[k,i]).  The 8.6 GFLOP A-GEMM +
// softmax cancels exactly; the kernel is pure bandwidth:
//   x(0.25MB) + W(64MB) + b(64MB) ~= 128.25MB -> ~5.5us @ 23.3 TB/s.
//
// U = x @ W via V_WMMA_F32_16X16X4_F32 (exact fp32), M=batch(8 real rows),
// N=D1 tile(2x16), K=D0 in 4 steps.  W tiles are streamed through LDS with
// GLOBAL_LOAD_ASYNC_TO_LDS_B128 (double-buffered per wave, ASYNCcnt-paced),
// so the 64MB W stream is pure 16B-coalesced async traffic.
// ---------------------------------------------------------------------------

typedef __attribute__((ext_vector_type(2))) float v2f;
typedef __attribute__((ext_vector_type(8))) float v8f;

#define B_      8
#define N0_     512
#define D0_     16
#define N1_     64
#define D1_     32
#define NCHUNK  8
#define I_PER_CHUNK (N0_ / NCHUNK)   /* 64 */
#define TILES_PER_WAVE (I_PER_CHUNK / 4) /* 16 */
#define TILE_F  (D0_ * D1_)          /* 512 floats = 2KB */
#define EPS_    1e-5f

// d_ws layout (floats):
//   [0, N1_*N0_)                  : wsum[k*N0_+i] = 1 + sum_z bias[k,i,z]
//   [N1_*N0_, +N1_*NCHUNK*B_*D1_) : partialS[k][chunk][b][l]

// ---- Kernel 1: row sums of bias (one wave per 512-wide row) ---------------
__global__ __launch_bounds__(256) void caps_bsum_kernel(
    const float* __restrict__ bias, float* __restrict__ wsum) {
  const int wave = threadIdx.x >> 5;
  const int lane = threadIdx.x & 31;
  const int row  = blockIdx.x * 8 + wave;              // [0, N1_*N0_)
  const float4* p = reinterpret_cast<const float4*>(bias + (size_t)row * N0_);
  float s = 0.f;
#pragma unroll
  for (int q = 0; q < 4; ++q) {                        // 4 * 32 lanes * float4 = 512
    float4 v = p[lane + 32 * q];
    s += (v.x + v.y) + (v.z + v.w);
  }
#pragma unroll
  for (int m = 16; m >= 1; m >>= 1) s += __shfl_xor(s, m, 32);
  if (lane == 0) wsum[row] = 1.0f + s;
}

// ---- async copy of one 2KB W tile (4 x b128 x 32 lanes) -------------------
// CDNA5 ISA: each lane copies 16B from its global to its LDS address;
// INST_OFFSET advances both sides, so one base covers the whole tile.
__device__ __forceinline__ void async_tile_2k(unsigned lds_byte, const float* g) {
  asm volatile("global_load_async_to_lds_b128 %0, %1, off"
               :: "v"(lds_byte), "v"(g) : "memory");
  asm volatile("global_load_async_to_lds_b128 %0, %1, off offset:512"
               :: "v"(lds_byte), "v"(g) : "memory");
  asm volatile("global_load_async_to_lds_b128 %0, %1, off offset:1024"
               :: "v"(lds_byte), "v"(g) : "memory");
  asm volatile("global_load_async_to_lds_b128 %0, %1, off offset:1536"
               :: "v"(lds_byte), "v"(g) : "memory");
}

// ---- Kernel 2: U via f32 WMMA, async-LDS W staging, fused i-accumulation --
__global__ __launch_bounds__(128) void caps_u_s_kernel(
    const float* __restrict__ x, const float* __restrict__ W,
    const float* __restrict__ wsum, float* __restrict__ partialS) {
  const int chunk = blockIdx.x;
  const int k     = blockIdx.y;
  const int wave  = threadIdx.x >> 5;
  const int lane  = threadIdx.x & 31;
  // ISA 7.12.2 operand striping (wave32):
  //   A 16x4 f32 : lane holds row M=lane%16; lanes<16 -> K {0,1}, lanes>=16 -> {2,3}
  //   B 4x16 f32 : lane holds col N=lane%16; same K-pair striping
  //   D 16x16 f32: VGPR v, lanes<16 -> M=v (batch b), N=lane
  const int M   = lane & 15;
  const int klo = (lane >> 4) << 1;

  __shared__ float Wbuf[4][2][TILE_F];   // per-wave double buffer, 16KB
  __shared__ float Sred[4][2][8][16];    // cross-wave S reduction, 4KB

  float S0[8], S1[8];
#pragma unroll
  for (int v = 0; v < 8; ++v) { S0[v] = 0.f; S1[v] = 0.f; }

  const int i0 = chunk * I_PER_CHUNK + wave;           // this wave: i = i0 + 4t
  const float* Wt = W + (size_t)(k * N0_ + i0) * TILE_F;
  const unsigned ldsb[2] = {
      (unsigned)(uintptr_t)(&Wbuf[wave][0][0]) + (unsigned)lane * 16u,
      (unsigned)(uintptr_t)(&Wbuf[wave][1][0]) + (unsigned)lane * 16u };

  async_tile_2k(ldsb[0], Wt + lane * 4);               // prologue: tile 0 -> buf 0
  int buf = 0;

  for (int t = 0; t < TILES_PER_WAVE; ++t) {
    const int i = i0 + 4 * t;
    if (t + 1 < TILES_PER_WAVE) {                      // issue next tile, then pace
      async_tile_2k(ldsb[buf ^ 1], Wt + 4 * TILE_F + lane * 4);
      asm volatile("s_wait_asynccnt 0x4" ::: "memory"); // current tile's 4 retired
    } else {
      asm volatile("s_wait_asynccnt 0x0" ::: "memory");
    }

    const float w = wsum[k * N0_ + i];                 // 1 + bsum[k,i]
    const float* xr = x + ((size_t)M * N0_ + i) * D0_; // valid only when M < 8
    const float* Wb = &Wbuf[wave][buf][0];             // LDS tile (j-major, 32 l)

    v8f acc0 = {0.f,0.f,0.f,0.f,0.f,0.f,0.f,0.f};
    v8f acc1 = acc0;
#pragma unroll
    for (int s = 0; s < 4; ++s) {                      // K = D0 in 4 steps of 4
      const int j0 = s * 4 + klo;
      float ax = 0.f, ay = 0.f;
      if (M < B_) {                                    // A rows 8..15 are zero pad
        float2 xv = *reinterpret_cast<const float2*>(xr + j0);
        ax = xv.x; ay = xv.y;
      }
      v2f a;  a[0] = ax;                        a[1] = ay;
      v2f b0; b0[0] = Wb[j0 * D1_ + M];         b0[1] = Wb[(j0 + 1) * D1_ + M];
      v2f b1; b1[0] = Wb[j0 * D1_ + 16 + M];    b1[1] = Wb[(j0 + 1) * D1_ + 16 + M];
      acc0 = __builtin_amdgcn_wmma_f32_16x16x4_f32(false, a, false, b0,
                                                   (short)0, acc0, false, false);
      acc1 = __builtin_amdgcn_wmma_f32_16x16x4_f32(false, a, false, b1,
                                                   (short)0, acc1, false, false);
    }
    // lanes<16: acc0[v] = U[b=v, i, lane], acc1[v] = U[b=v, i, 16+lane]
#pragma unroll
    for (int v = 0; v < 8; ++v) { S0[v] += w * acc0[v]; S1[v] += w * acc1[v]; }

    Wt += 4 * TILE_F;
    buf ^= 1;
  }

  if (lane < 16) {
#pragma unroll
    for (int v = 0; v < 8; ++v) {
      Sred[wave][0][v][lane] = S0[v];
      Sred[wave][1][v][lane] = S1[v];
    }
  }
  __syncthreads();
  for (int idx = threadIdx.x; idx < B_ * D1_; idx += blockDim.x) {
    const int b = idx >> 5, l = idx & 31;
    const int t = l >> 4, ll = l & 15;
    float s = Sred[0][t][b][ll] + Sred[1][t][b][ll]
            + Sred[2][t][b][ll] + Sred[3][t][b][ll];
    partialS[(((size_t)k * NCHUNK + chunk) * B_ + b) * D1_ + l] = s;
  }
}

// ---- Kernel 3: reduce chunks + squash -------------------------------------
__global__ __launch_bounds__(256) void caps_reduce_squash_kernel(
    const float* __restrict__ partialS, float* __restrict__ out) {
  const int k = blockIdx.x;
  const int b = threadIdx.x >> 5;
  const int l = threadIdx.x & 31;
  float s = 0.f;
#pragma unroll
  for (int c = 0; c < NCHUNK; ++c)
    s += partialS[(((size_t)k * NCHUNK + c) * B_ + b) * D1_ + l];
  __shared__ float Sf[B_][D1_];
  Sf[b][l] = s;
  __syncthreads();
  float sq = 0.f;
#pragma unroll
  for (int ll = 0; ll < D1_; ++ll) sq += Sf[b][ll] * Sf[b][ll];
  const float scale = (sq / (1.f + sq)) / (sqrtf(sq) + EPS_);
  out[((size_t)b * N1_ + k) * D1_ + l] = scale * s;
}

// ---------------------------------------------------------------------------
extern "C" void kernel_launch(void* const* d_in, const int* in_sizes, int n_in,
                              void* d_out, int out_size, void* d_ws, size_t ws_size,
                              hipStream_t stream) {
  (void)in_sizes; (void)n_in; (void)out_size; (void)ws_size;
  const float* x    = (const float*)d_in[0];   // [8,512,16]
  const float* W    = (const float*)d_in[1];   // [64,512,16,32]
  const float* bias = (const float*)d_in[2];   // [64,512,512]
  float* out  = (float*)d_out;                 // [8,64,32]
  float* wsum = (float*)d_ws;                  // 32768 floats
  float* partialS = wsum + (size_t)N1_ * N0_;  // 131072 floats

  caps_bsum_kernel<<<dim3((N1_ * N0_) / 8), dim3(256), 0, stream>>>(bias, wsum);
  caps_u_s_kernel<<<dim3(NCHUNK, N1_), dim3(128), 0, stream>>>(x, W, wsum, partialS);
  caps_reduce_squash_kernel<<<dim3(N1_), dim3(256), 0, stream>>>(partialS, out);
}